// CosineLSM_47236050321596
// MI455X (gfx1250) — compile-verified
//
#include <hip/hip_runtime.h>
#include <hip/hip_bf16.h>

typedef __attribute__((ext_vector_type(2))) float v2f;
typedef __attribute__((ext_vector_type(8))) float v8f;

#define IN_DIM 128        // input features
#define OUT_DIM 128       // M * HID = 2 * 64
#define HEADS 2
#define COS_EPS 1e-8f

// ---------------------------------------------------------------------------
// Kernel 1: h = relu(x @ W + b) using native fp32 WMMA 16x16x4.
// grid.x = ceil(N/16) row tiles; 8 waves per block, one 16-col tile per wave.
// A fragment (16x4 f32): lanes 0-15 hold rows M=0..15 with K=k0,k0+1 in v0,v1;
// lanes 16-31 hold the same rows with K=k0+2,k0+3. B (4x16) mirrored on cols.
// C/D (16x16 f32): VGPR j -> row M=j (lanes 0-15) / M=j+8 (lanes 16-31), N=lane&15.
// ---------------------------------------------------------------------------
__global__ void encoder_gemm(const float* __restrict__ x,
                             const float* __restrict__ W,
                             const float* __restrict__ bias,
                             float* __restrict__ h,
                             int N) {
    const int lane   = threadIdx.x & 31;
    const int wave   = threadIdx.x >> 5;        // 0..7 -> column tile
    const int l15    = lane & 15;
    const int half   = lane >> 4;               // 0: K pair 0, 1: K pair 2
    const int rowBase = blockIdx.x * 16;
    const int colBase = wave * 16;

    // clamp load row so EXEC can stay all-ones for WMMA on a ragged edge
    int arow = rowBase + l15;
    if (arow >= N) arow = N - 1;
    const int bcol = colBase + l15;

    v8f acc = {};
    #pragma unroll
    for (int k0 = 0; k0 < IN_DIM; k0 += 4) {
        const int kk = k0 + half * 2;                       // even -> 8B aligned
        v2f A = *(const v2f*)(x + (size_t)arow * IN_DIM + kk);
        v2f B;
        B.x = W[(size_t)kk * OUT_DIM + bcol];
        B.y = W[(size_t)(kk + 1) * OUT_DIM + bcol];
        acc = __builtin_amdgcn_wmma_f32_16x16x4_f32(
            /*neg_a=*/false, A, /*neg_b=*/false, B,
            /*c_mod=*/(short)0, acc, /*reuse_a=*/false, /*reuse_b=*/false);
    }

    const float bv = bias[bcol];
    #pragma unroll
    for (int j = 0; j < 8; ++j) {
        const int row = rowBase + j + half * 8;
        if (row < N) {
            float v = acc[j] + bv;
            h[(size_t)row * OUT_DIM + bcol] = v > 0.0f ? v : 0.0f;
        }
    }
}

// ---------------------------------------------------------------------------
// Kernel 2: per-node, per-head L2 normalization in place.
// One wave per node. Lane l holds float4 l of the 128-float row; lanes 0-15
// cover head 0 (64 floats), lanes 16-31 cover head 1. Norm clamped at eps.
// ---------------------------------------------------------------------------
__global__ void normalize_nodes(float* __restrict__ h, int N) {
    const int node = blockIdx.x * (blockDim.x >> 5) + (threadIdx.x >> 5);
    if (node >= N) return;
    const int lane = threadIdx.x & 31;

    float4* row = (float4*)(h + (size_t)node * OUT_DIM);
    float4 v = row[lane];
    float ss = v.x * v.x + v.y * v.y + v.z * v.z + v.w * v.w;
    // reduce within each 16-lane half (one head each)
    ss += __shfl_xor(ss, 1, 32);
    ss += __shfl_xor(ss, 2, 32);
    ss += __shfl_xor(ss, 4, 32);
    ss += __shfl_xor(ss, 8, 32);
    const float inv = 1.0f / fmaxf(sqrtf(ss), COS_EPS);
    v.x *= inv; v.y *= inv; v.z *= inv; v.w *= inv;
    row[lane] = v;
}

// ---------------------------------------------------------------------------
// Kernel 3: per-edge cosine. One wave per edge; 512B coalesced row gathers
// (L2-resident), per-head dot via half-wave reduction, heads combined by
// xor-16. out = (c0+c1)/4 + 0.5 + 1e-4 == mean over heads of (cos+1)/2+1e-4.
// ---------------------------------------------------------------------------
__global__ void edge_cosine(const float* __restrict__ hn,
                            const long long* __restrict__ ei,
                            float* __restrict__ out,
                            int E) {
    const int e = blockIdx.x * (blockDim.x >> 5) + (threadIdx.x >> 5);
    if (e >= E) return;
    const int lane = threadIdx.x & 31;

    const long long s = ei[e];
    const long long d = ei[(long long)E + e];
    const float4 q = ((const float4*)(hn + s * OUT_DIM))[lane];
    const float4 k = ((const float4*)(hn + d * OUT_DIM))[lane];

    float p = q.x * k.x + q.y * k.y + q.z * k.z + q.w * k.w;
    p += __shfl_xor(p, 1, 32);
    p += __shfl_xor(p, 2, 32);
    p += __shfl_xor(p, 4, 32);
    p += __shfl_xor(p, 8, 32);   // lower half: c0 (head 0), upper half: c1
    p += __shfl_xor(p, 16, 32);  // all lanes: c0 + c1

    if (lane == 0) out[e] = 0.25f * p + 0.5f + 1e-4f;
}

// ---------------------------------------------------------------------------
// inputs: 0=x [N,128] f32, 1=edge_index [2,E] i64, 2=batch [N] i64 (unused),
//         3=W [128,128] f32, 4=b [128] f32.   out: [E] f32.   ws: h buffer.
// ---------------------------------------------------------------------------
extern "C" void kernel_launch(void* const* d_in, const int* in_sizes, int n_in,
                              void* d_out, int out_size, void* d_ws, size_t ws_size,
                              hipStream_t stream) {
    const float*     x  = (const float*)d_in[0];
    const long long* ei = (const long long*)d_in[1];
    const float*     W  = (const float*)d_in[3];
    const float*     b  = (const float*)d_in[4];
    float* out = (float*)d_out;
    float* h   = (float*)d_ws;                 // N*128 floats of scratch

    const int N = in_sizes[0] / IN_DIM;
    const int E = in_sizes[1] / 2;

    // 1) encoder GEMM (WMMA f32), bias + relu fused
    {
        dim3 block(256);                       // 8 waves: 8 column tiles of 16
        dim3 grid((N + 15) / 16);
        encoder_gemm<<<grid, block, 0, stream>>>(x, W, b, h, N);
    }
    // 2) per-node per-head normalization (in place)
    {
        dim3 block(256);                       // 8 nodes per block
        dim3 grid((N + 7) / 8);
        normalize_nodes<<<grid, block, 0, stream>>>(h, N);
    }
    // 3) per-edge cosine similarity
    {
        dim3 block(256);                       // 8 edges per block
        dim3 grid((E + 7) / 8);
        edge_cosine<<<grid, block, 0, stream>>>(h, ei, out, E);
    }
}